// GCNModel_7902739825366
// MI455X (gfx1250) — compile-verified
//
#include <hip/hip_runtime.h>

typedef __attribute__((ext_vector_type(2))) float v2f;
typedef __attribute__((ext_vector_type(8))) float v8f;

#define IN_FEAT 512
#define H1F 8
#define H2F 4
#define NCLSF 2

// ---------------------------------------------------------------------------
// Degree accumulation: deg[dst] += ew  (self-loop +1 folded into dinv kernel)
// ---------------------------------------------------------------------------
__global__ void deg_kernel(const long long* __restrict__ dst,
                           const float* __restrict__ ew,
                           float* __restrict__ deg, int n_edges) {
  int e = blockIdx.x * blockDim.x + threadIdx.x;
  if (e < n_edges) atomicAdd(&deg[(int)dst[e]], ew[e]);
}

__global__ void dinv_kernel(const float* __restrict__ deg,
                            float* __restrict__ dinv, int n) {
  int i = blockIdx.x * blockDim.x + threadIdx.x;
  if (i < n) dinv[i] = rsqrtf(deg[i] + 1.0f);   // deg+1 >= 1, always valid
}

// ---------------------------------------------------------------------------
// Layer-1 GEMM: h1[100000 x 8] = x[100000 x 512] @ W1[512 x 8]
// One wave per 16-row tile, V_WMMA_F32_16X16X4_F32, 128 k-steps.
// B is staged in LDS pre-swizzled into the exact WMMA B-fragment layout and
// zero-padded to N=16, so the hot loop is branch-free:
//   global_load_b64 (A stream) + ds_load_b64 (B frag) + v_wmma.
// ---------------------------------------------------------------------------
__global__ void __launch_bounds__(128) gemm1_wmma(const float* __restrict__ x,
                                                  const float* __restrict__ W1,
                                                  float* __restrict__ h1,
                                                  int n_nodes) {
  // sB[g*32 + l] = {W1[4g+kh][nn], W1[4g+kh+1][nn]},  kh=(l>>4)*2, nn=l&15
  // (zero for padded cols nn >= 8).  128 groups * 32 lanes * 8B = 32 KB.
  __shared__ v2f sB[(IN_FEAT / 4) * 32];
  for (int idx = threadIdx.x; idx < (IN_FEAT / 4) * 32; idx += 128) {
    int l  = idx & 31;
    int g  = idx >> 5;
    int kh = (l >> 4) << 1;
    int nn = l & 15;
    v2f b = {0.0f, 0.0f};
    if (nn < H1F) {
      b.x = W1[(4 * g + kh) * H1F + nn];
      b.y = W1[(4 * g + kh + 1) * H1F + nn];
    }
    sB[idx] = b;
  }
  __syncthreads();

  const int lane = threadIdx.x & 31;
  const int tile = blockIdx.x * 4 + (threadIdx.x >> 5);   // wave-uniform
  if (tile * 16 >= n_nodes) return;                       // whole wave exits
  const int row0 = tile * 16;

  // A 16x4 f32 layout: lanes 0-15 -> M=lane, K0/K1; lanes 16-31 -> K2/K3
  const int m  = lane & 15;
  const int kh = (lane >> 4) << 1;
  const float* xrow = x + (size_t)(row0 + m) * IN_FEAT + kh;
  const v2f* bcol = sB + lane;

  v8f acc = {};
#pragma unroll 8
  for (int g = 0; g < IN_FEAT / 4; ++g) {
    v2f a = *reinterpret_cast<const v2f*>(xrow + 4 * g);  // 8B coalesced
    v2f b = bcol[(size_t)g * 32];                         // ds_load_b64, imm offset
    acc = __builtin_amdgcn_wmma_f32_16x16x4_f32(
        /*neg_a=*/false, a, /*neg_b=*/false, b,
        /*c_mod=*/(short)0, acc, /*reuse_a=*/false, /*reuse_b=*/false);
  }

  // C/D layout: VGPR r, lanes 0-15 -> M=r; lanes 16-31 -> M=r+8; N = lane&15
  const int nn = lane & 15;
  if (nn < H1F) {
    const int mBase = row0 + ((lane >> 4) << 3);
#pragma unroll
    for (int r = 0; r < 8; ++r)
      h1[(size_t)(mBase + r) * H1F + nn] = acc[r];
  }
}

// ---------------------------------------------------------------------------
// Edge aggregation: agg[dst] += dinv[src]*ew*dinv[dst] * h[src]   (F = 8/4/2)
// Thread-per-edge; feature loop unrolled so the h[src] gather merges to b128.
// Edge arrays (64 MB) are L2-resident (192 MB) after the first pass.
// ---------------------------------------------------------------------------
template <int F>
__global__ void aggregate_kernel(const long long* __restrict__ src,
                                 const long long* __restrict__ dst,
                                 const float* __restrict__ ew,
                                 const float* __restrict__ dinv,
                                 const float* __restrict__ h,
                                 float* __restrict__ agg, int n_edges) {
  int e = blockIdx.x * blockDim.x + threadIdx.x;
  if (e >= n_edges) return;
  int s = (int)src[e];
  int d = (int)dst[e];
  float norm = dinv[s] * ew[e] * dinv[d];
  const float* hs = h + (size_t)s * F;
  float* ad = agg + (size_t)d * F;
#pragma unroll
  for (int f = 0; f < F; ++f) atomicAdd(&ad[f], norm * hs[f]);
}

// ---------------------------------------------------------------------------
// Self-loop + bias (+ optional ReLU): agg = [relu](agg + dinv^2 * h + b)
// ---------------------------------------------------------------------------
template <int F, bool RELU>
__global__ void finalize_kernel(const float* __restrict__ h,
                                const float* __restrict__ dinv,
                                const float* __restrict__ bias,
                                float* __restrict__ agg, int n_nodes) {
  int i = blockIdx.x * blockDim.x + threadIdx.x;
  if (i >= n_nodes) return;
  float di2 = dinv[i] * dinv[i];
#pragma unroll
  for (int f = 0; f < F; ++f) {
    float v = agg[(size_t)i * F + f] + di2 * h[(size_t)i * F + f] + bias[f];
    agg[(size_t)i * F + f] = RELU ? fmaxf(v, 0.0f) : v;
  }
}

// ---------------------------------------------------------------------------
// Tiny per-node dense layers (8->4, 4->2): scalar FMAs, bandwidth-trivial.
// ---------------------------------------------------------------------------
template <int IN, int OUT>
__global__ void dense_kernel(const float* __restrict__ g,
                             const float* __restrict__ W,
                             float* __restrict__ h, int n_nodes) {
  int i = blockIdx.x * blockDim.x + threadIdx.x;
  if (i >= n_nodes) return;
  float in[IN];
#pragma unroll
  for (int k = 0; k < IN; ++k) in[k] = g[(size_t)i * IN + k];
#pragma unroll
  for (int o = 0; o < OUT; ++o) {
    float acc = 0.0f;
#pragma unroll
    for (int k = 0; k < IN; ++k) acc = fmaf(in[k], W[k * OUT + o], acc);
    h[(size_t)i * OUT + o] = acc;
  }
}

// ---------------------------------------------------------------------------
extern "C" void kernel_launch(void* const* d_in, const int* in_sizes, int n_in,
                              void* d_out, int out_size, void* d_ws, size_t ws_size,
                              hipStream_t stream) {
  const float*     x    = (const float*)d_in[0];
  const long long* eidx = (const long long*)d_in[1];   // int64 [2, E]
  const float*     ew   = (const float*)d_in[2];
  const float*     W1   = (const float*)d_in[3];
  const float*     b1   = (const float*)d_in[4];
  const float*     W2   = (const float*)d_in[5];
  const float*     b2   = (const float*)d_in[6];
  const float*     W3   = (const float*)d_in[7];
  const float*     b3   = (const float*)d_in[8];

  const int n_nodes = in_sizes[0] / IN_FEAT;
  const int n_edges = in_sizes[2];
  const long long* src = eidx;
  const long long* dst = eidx + n_edges;

  // Workspace layout (floats): deg | dinv | h1(8) | g1(8) | h2(4) | g2(4) | h3(2)
  float* ws   = (float*)d_ws;
  float* deg  = ws;
  float* dinv = ws + (size_t)1  * n_nodes;
  float* h1   = ws + (size_t)2  * n_nodes;
  float* g1   = ws + (size_t)10 * n_nodes;
  float* h2   = ws + (size_t)18 * n_nodes;
  float* g2   = ws + (size_t)22 * n_nodes;
  float* h3   = ws + (size_t)26 * n_nodes;
  float* outp = (float*)d_out;

  // Zero the atomic-accumulation targets (capture-safe memset nodes).
  hipMemsetAsync(deg,  0, (size_t)n_nodes * sizeof(float), stream);
  hipMemsetAsync(g1,   0, (size_t)n_nodes * H1F  * sizeof(float), stream);
  hipMemsetAsync(g2,   0, (size_t)n_nodes * H2F  * sizeof(float), stream);
  hipMemsetAsync(outp, 0, (size_t)n_nodes * NCLSF * sizeof(float), stream);

  const int TB = 256;
  const int ebl = (n_edges + TB - 1) / TB;
  const int nbl = (n_nodes + TB - 1) / TB;

  // Shared normalization (identical across all three GCN layers)
  deg_kernel<<<ebl, TB, 0, stream>>>(dst, ew, deg, n_edges);
  dinv_kernel<<<nbl, TB, 0, stream>>>(deg, dinv, n_nodes);

  // Layer 1: WMMA transform, then normalized aggregation + self-loop + ReLU
  const int tiles = (n_nodes + 15) / 16;
  gemm1_wmma<<<(tiles + 3) / 4, 128, 0, stream>>>(x, W1, h1, n_nodes);
  aggregate_kernel<H1F><<<ebl, TB, 0, stream>>>(src, dst, ew, dinv, h1, g1, n_edges);
  finalize_kernel<H1F, true><<<nbl, TB, 0, stream>>>(h1, dinv, b1, g1, n_nodes);

  // Layer 2
  dense_kernel<H1F, H2F><<<nbl, TB, 0, stream>>>(g1, W2, h2, n_nodes);
  aggregate_kernel<H2F><<<ebl, TB, 0, stream>>>(src, dst, ew, dinv, h2, g2, n_edges);
  finalize_kernel<H2F, true><<<nbl, TB, 0, stream>>>(h2, dinv, b2, g2, n_nodes);

  // Layer 3 (aggregates straight into d_out, no ReLU)
  dense_kernel<H2F, NCLSF><<<nbl, TB, 0, stream>>>(g2, W3, h3, n_nodes);
  aggregate_kernel<NCLSF><<<ebl, TB, 0, stream>>>(src, dst, ew, dinv, h3, outp, n_edges);
  finalize_kernel<NCLSF, false><<<nbl, TB, 0, stream>>>(h3, dinv, b3, outp, n_nodes);
}